// TensorizedEmbedding_78374563217858
// MI455X (gfx1250) — compile-verified
//
#include <hip/hip_runtime.h>
#include <stdint.h>

typedef __attribute__((ext_vector_type(2))) float v2f;
typedef __attribute__((ext_vector_type(8))) float v8f;

// TT shapes: VOC = 100*100*100, EMB = 4*4*8, RANK = 16
// core0: [100, 4, 16]      (leading rank-1 squeezed)
// core1: [16, 100, 4, 16]
// core2: [16, 100, 8]      (trailing rank-1 squeezed)

__global__ __launch_bounds__(256) void tt_embed_wmma_kernel(
    const int*   __restrict__ x,      // [nrows]
    const float* __restrict__ core0,  // [100,4,16]
    const float* __restrict__ core1,  // [16,100,4,16]
    const float* __restrict__ core2,  // [16,100,8]
    float*       __restrict__ out,    // [nrows,128]
    int nrows)
{
    // Per-wave scratch: 1024 f32 core1 slice + 128 f32 core2 slice
    __shared__ float lds[8][1152];

    const int lane = threadIdx.x & 31;
    const int wave = threadIdx.x >> 5;
    const int row  = blockIdx.x * 8 + wave;
    if (row >= nrows) return;                 // wave-uniform

    float* g1   = lds[wave];
    float* g2   = lds[wave] + 1024;
    float* orow = out + (size_t)row * 128;

    // idx is identical across the wave -> force scalar (saddr-form gathers).
    const int idx = __builtin_amdgcn_readfirstlane(x[row]);

    if (idx == 0) {                           // PAD row -> zeros (wave-uniform)
        float4 z = make_float4(0.f, 0.f, 0.f, 0.f);
        ((float4*)orow)[lane] = z;            // 32 lanes x 4 floats = 128
        return;
    }

    const int i1  = idx / 10000;
    const int rem = idx - i1 * 10000;
    const int i2  = rem / 100;
    const int i3  = rem - i2 * 100;

    // ---- Stage 0a: async-DMA core1[:, i2, :, :] (16x4x16 f32) into LDS.
    // Logical image g = r1*64 + (n2*16 + r2); lane covers [32*lane, 32*lane+32).
    // INST_OFFSET is added to both LDS and global addresses, and both advance
    // by 16B per chunk, so one address pair + offset:0..112 covers all 8.
    {
        const int r1 = lane >> 1;
        const int q0 = (lane & 1) * 32;
        unsigned long long ga =
            (unsigned long long)(uintptr_t)(core1 + (size_t)(r1 * 100 + i2) * 64 + q0);
        unsigned loff = (unsigned)(uintptr_t)(g1 + lane * 32);   // LDS byte offset
        asm volatile(
            "global_load_async_to_lds_b128 %0, %1, off\n\t"
            "global_load_async_to_lds_b128 %0, %1, off offset:16\n\t"
            "global_load_async_to_lds_b128 %0, %1, off offset:32\n\t"
            "global_load_async_to_lds_b128 %0, %1, off offset:48\n\t"
            "global_load_async_to_lds_b128 %0, %1, off offset:64\n\t"
            "global_load_async_to_lds_b128 %0, %1, off offset:80\n\t"
            "global_load_async_to_lds_b128 %0, %1, off offset:96\n\t"
            "global_load_async_to_lds_b128 %0, %1, off offset:112"
            :: "v"(loff), "v"(ga) : "memory");
    }
    // ---- Stage 0b: async-DMA core2[:, i3, :] (16x8 f32) into LDS.
    // Logical image h = r2*8 + n3; lane covers [4*lane, 4*lane+4).
    {
        const int r2 = lane >> 1;
        const int n3 = (lane & 1) * 4;
        unsigned long long ga =
            (unsigned long long)(uintptr_t)(core2 + (size_t)(r2 * 100 + i3) * 8 + n3);
        unsigned loff = (unsigned)(uintptr_t)(g2 + lane * 4);
        asm volatile("global_load_async_to_lds_b128 %0, %1, off"
                     :: "v"(loff), "v"(ga) : "memory");
    }

    // ---- Per-lane identity (A and B f32 operand layouts are lane-symmetric):
    // lane L owns non-K index L%16; hi = L/16 selects K pair {2hi, 2hi+1} per block.
    const int m  = lane & 15;        // T2 row (n1,n2) == D' column
    const int hi = lane >> 4;
    const int n1 = m >> 2;
    const int n2 = m & 3;

    // T1 row for this lane's n1: core0[i1, n1, 0..15] (normal loads, L2-resident)
    float t1[16];
    {
        const float4* s4 = (const float4*)(core0 + (size_t)(i1 * 4 + n1) * 16);
        #pragma unroll
        for (int v = 0; v < 4; ++v) {
            float4 f = s4[v];
            t1[4*v+0] = f.x; t1[4*v+1] = f.y; t1[4*v+2] = f.z; t1[4*v+3] = f.w;
        }
    }

    // Wait for the async LDS fills before any ds_load consumes them.
    asm volatile("s_wait_asynccnt 0x0" ::: "memory");

    // ---- Stage 1 (packed VALU): T2[m, k] = sum_{r1} t1[r1] * G1[r1, n2, k]
    // acc[b] = { T2[m, 4b+2hi], T2[m, 4b+2hi+1] }  == WMMA B' block b (T2^T)
    v2f acc[4];
    #pragma unroll
    for (int b = 0; b < 4; ++b) { acc[b].x = 0.f; acc[b].y = 0.f; }

    const float* g1base = g1 + n2 * 16 + hi * 2;   // + r1*64 + 4*b
    #pragma unroll
    for (int r1 = 0; r1 < 16; ++r1) {
        const float t = t1[r1];
        const float* p = g1base + r1 * 64;
        #pragma unroll
        for (int b = 0; b < 4; ++b) {
            v2f g = *(const v2f*)(p + 4 * b);      // 8B-aligned ds_load_b64
            acc[b].x += t * g.x;
            acc[b].y += t * g.y;
        }
    }

    // ---- A' operand = G2^T: lane L holds A'[n3 = L%16, k in {4b+2hi,+1}] =
    // G2[k, L%16]. Rows n3>=8 are don't-care (their D' rows live only in
    // lanes 16-31, never stored) -> clamp with m&7, no divergence.
    const int mc = m & 7;
    v2f ag[4];
    #pragma unroll
    for (int b = 0; b < 4; ++b) {
        const int k0 = 4 * b + 2 * hi;
        ag[b].x = g2[k0 * 8 + mc];
        ag[b].y = g2[(k0 + 1) * 8 + mc];
    }

    // ---- Stage 2 (WMMA, transposed): D'[n3, m] = sum_{r2} G2[r2,n3] * T2[m,r2]
    // = out[m*8 + n3].  Four chained V_WMMA_F32_16X16X4_F32.
    v8f c = {};
    #pragma unroll
    for (int b = 0; b < 4; ++b) {
        c = __builtin_amdgcn_wmma_f32_16x16x4_f32(
                /*neg_a=*/false, ag[b], /*neg_b=*/false, acc[b],
                /*c_mod=*/(short)0, c, /*reuse_a=*/false, /*reuse_b=*/false);
    }

    // ---- Store: lane L<16, VGPR j holds out[L*8 + j] -> 8 contiguous floats
    // per lane; two coalesced 128-bit stores cover the whole row.
    if (lane < 16) {
        float4* dst = (float4*)(orow + lane * 8);
        dst[0] = make_float4(c[0], c[1], c[2], c[3]);
        dst[1] = make_float4(c[4], c[5], c[6], c[7]);
    }
}

extern "C" void kernel_launch(void* const* d_in, const int* in_sizes, int n_in,
                              void* d_out, int out_size, void* d_ws, size_t ws_size,
                              hipStream_t stream) {
    const int*   x  = (const int*)  d_in[0];   // x (flattened 64*512)
    const float* c0 = (const float*)d_in[1];   // core0 (1,100,4,16)
    const float* c1 = (const float*)d_in[2];   // core1 (16,100,4,16)
    const float* c2 = (const float*)d_in[3];   // core2 (16,100,8,1)
    float* out = (float*)d_out;

    const int nrows  = in_sizes[0];            // 32768
    const int blocks = (nrows + 7) / 8;        // 8 rows (one per wave) per block

    tt_embed_wmma_kernel<<<blocks, 256, 0, stream>>>(x, c0, c1, c2, out, nrows);
}